// SinkhornDistance_87024627351930
// MI455X (gfx1250) — compile-verified
//
#include <hip/hip_runtime.h>

// Sinkhorn distance, MI455X (gfx1250, wave32).
// Phase 1: bandwidth-bound 32x32x524288 fp32 GEMM + row norms via
//          V_WMMA_F32_16X16X4_F32 (fp32 in / fp32 acc).
//          Data movement: GLOBAL_LOAD_ASYNC_TO_LDS_B128 double-buffered
//          (ASYNCcnt-tracked), falling back to register staging if the
//          async builtins are unavailable.
// Phase 2: single-workgroup 100-iteration Sinkhorn loop on the 32x32 matrix.

#define DFEAT 524288
#define NPTS 32
#define EPS_F 0.001f
#define INV_EPS 1000.0f
#define MAX_ITER_S 100
#define GBLK 256            // phase-1 blocks (K-slices)
#define PART_STRIDE 1088    // per-block partial: 1024 (xy tile) + 32 (x2) + 32 (y2)

#if defined(__gfx1250__) && __has_builtin(__builtin_amdgcn_global_load_async_to_lds_b128) && __has_builtin(__builtin_amdgcn_s_wait_asynccnt)
#define USE_ASYNC 1
#else
#define USE_ASYNC 0
#endif

typedef __attribute__((ext_vector_type(2))) float v2f;
typedef __attribute__((ext_vector_type(8))) float v8f;

__device__ __forceinline__ float wave_reduce_max(float v) {
#pragma unroll
  for (int off = 16; off > 0; off >>= 1) v = fmaxf(v, __shfl_xor(v, off, 32));
  return v;
}
__device__ __forceinline__ float wave_reduce_sum(float v) {
#pragma unroll
  for (int off = 16; off > 0; off >>= 1) v += __shfl_xor(v, off, 32);
  return v;
}

#if USE_ASYNC
// Builtin signature (from hipcc diagnostic): param 0 is a pointer to a
// 16-byte int vector in AS1 (printed as "int __vector_size__(16) __device__*").
typedef int v4i_vs __attribute__((__vector_size__(4 * sizeof(int))));
typedef __attribute__((address_space(1))) v4i_vs* gptr_b128;
typedef __attribute__((address_space(3))) v4i_vs* lptr_b128;

// Issue an async 32B (2 x b128) global->LDS copy for this lane.
// Generic->AS3 is a low-32-bit truncation on amdgcn; AS1 shares the flat value.
__device__ __forceinline__ void async_copy32B(const float* g, float* l) {
  gptr_b128 gp = (gptr_b128)(uintptr_t)g;
  lptr_b128 lp = (lptr_b128)(uint32_t)(uintptr_t)l;
  __builtin_amdgcn_global_load_async_to_lds_b128(gp, lp, 0, 0);
  __builtin_amdgcn_global_load_async_to_lds_b128(gp, lp, 16, 0);
}
#endif

// ---------------------------------------------------------------------------
// Phase 1: each block reduces a 2048-wide K-slice into a 32x32 partial xy tile
// plus partial row norms, written (plain stores, deterministic) to ws.
// ---------------------------------------------------------------------------
__global__ __launch_bounds__(256)
void sinkhorn_cost_partial(const float* __restrict__ x, const float* __restrict__ y,
                           float* __restrict__ ws) {
  // Row stride 68: 16B-aligned rows for b128 stores, bank = (4*row + col) % 64
  // -> conflict-free b64 fragment loads across the wave.
  __shared__ float sx[2][32 * 68];
  __shared__ float sy[2][32 * 68];
  __shared__ float sC[NPTS * NPTS];
  __shared__ float xn[NPTS];
  __shared__ float yn[NPTS];

  const int t    = threadIdx.x;
  const int r    = t >> 3;          // staging row (0..31), fixed per thread
  const int cg   = t & 7;           // staging col group
  const int lane = t & 31;
  const int w    = t >> 5;          // wave 0..7: owns K-subrange [w*8, w*8+8) of chunk
  const int KB   = DFEAT / GBLK;    // 2048 columns per block
  const long k0  = (long)blockIdx.x * KB;

  const float* gx = x + (size_t)r * DFEAT + k0 + cg * 8;
  const float* gy = y + (size_t)r * DFEAT + k0 + cg * 8;
  const int soff  = r * 68 + cg * 8;

  v8f acc00 = {}, acc01 = {}, acc10 = {}, acc11 = {};
  float ax_lo = 0.f, ax_hi = 0.f, ay_lo = 0.f, ay_hi = 0.f;  // fragment-based norms

  const int m = lane & 15;          // M (A) / N (B) index within 16-wide tile
  const int h = lane >> 4;          // half-wave selects K-pair 0/1 vs 2/3

#if USE_ASYNC
  // prime buffer 0
  async_copy32B(gx, &sx[0][soff]);
  async_copy32B(gy, &sy[0][soff]);
#else
  const float4* px = (const float4*)gx;
  const float4* py = (const float4*)gy;
  float4 rx0 = px[0], rx1 = px[1], ry0 = py[0], ry1 = py[1];
#endif

  int it = 0;
  for (int c = 0; c < KB; c += 64, ++it) {
    const int cur = it & 1;
#if USE_ASYNC
    if (c + 64 < KB) {
      // buf cur^1 was last read in the previous iteration (barrier-protected): safe.
      async_copy32B(gx + c + 64, &sx[cur ^ 1][soff]);
      async_copy32B(gy + c + 64, &sy[cur ^ 1][soff]);
      __builtin_amdgcn_s_wait_asynccnt(4);   // chunk c complete (in-order)
    } else {
      __builtin_amdgcn_s_wait_asynccnt(0);
    }
    __syncthreads();
    const float* bx = sx[cur];
    const float* by = sy[cur];
#else
    (void)cur;
    *(float4*)&sx[0][soff]     = rx0;
    *(float4*)&sx[0][soff + 4] = rx1;
    *(float4*)&sy[0][soff]     = ry0;
    *(float4*)&sy[0][soff + 4] = ry1;
    __syncthreads();
    if (c + 64 < KB) {
      const int o = (c + 64) >> 2;
      rx0 = px[o]; rx1 = px[o + 1];
      ry0 = py[o]; ry1 = py[o + 1];
    }
    const float* bx = sx[0];
    const float* by = sy[0];
#endif

#pragma unroll
    for (int s = 0; s < 2; ++s) {
      const int kk = w * 8 + s * 4 + 2 * h;   // per-lane K base (ISA A/B layout)
      v2f a0 = { bx[m * 68 + kk],        bx[m * 68 + kk + 1] };
      v2f a1 = { bx[(m + 16) * 68 + kk], bx[(m + 16) * 68 + kk + 1] };
      v2f b0 = { by[m * 68 + kk],        by[m * 68 + kk + 1] };
      v2f b1 = { by[(m + 16) * 68 + kk], by[(m + 16) * 68 + kk + 1] };
      // each (row,k) element lives in exactly one lane's fragment -> free norms
      ax_lo += a0[0] * a0[0] + a0[1] * a0[1];
      ax_hi += a1[0] * a1[0] + a1[1] * a1[1];
      ay_lo += b0[0] * b0[0] + b0[1] * b0[1];
      ay_hi += b1[0] * b1[0] + b1[1] * b1[1];
      acc00 = __builtin_amdgcn_wmma_f32_16x16x4_f32(false, a0, false, b0, (short)0, acc00, false, false);
      acc01 = __builtin_amdgcn_wmma_f32_16x16x4_f32(false, a0, false, b1, (short)0, acc01, false, false);
      acc10 = __builtin_amdgcn_wmma_f32_16x16x4_f32(false, a1, false, b0, (short)0, acc10, false, false);
      acc11 = __builtin_amdgcn_wmma_f32_16x16x4_f32(false, a1, false, b1, (short)0, acc11, false, false);
    }
    __syncthreads();
  }

  // Combine the 8 waves' accumulators in LDS (ds_add_f32), then store partial.
  for (int idx = t; idx < NPTS * NPTS; idx += 256) sC[idx] = 0.f;
  if (t < NPTS) { xn[t] = 0.f; yn[t] = 0.f; }
  __syncthreads();

  atomicAdd(&xn[m],      ax_lo);
  atomicAdd(&xn[m + 16], ax_hi);
  atomicAdd(&yn[m],      ay_lo);
  atomicAdd(&yn[m + 16], ay_hi);
#pragma unroll
  for (int vg = 0; vg < 8; ++vg) {
    const int M0 = vg + 8 * h;     // C/D layout: VGPR v, lanes16-31 -> M = v+8
    atomicAdd(&sC[M0 * 32 + m],             acc00[vg]);
    atomicAdd(&sC[M0 * 32 + 16 + m],        acc01[vg]);
    atomicAdd(&sC[(M0 + 16) * 32 + m],      acc10[vg]);
    atomicAdd(&sC[(M0 + 16) * 32 + 16 + m], acc11[vg]);
  }
  __syncthreads();

  float* outp = ws + (size_t)blockIdx.x * PART_STRIDE;
  for (int idx = t; idx < NPTS * NPTS; idx += 256) outp[idx] = sC[idx];
  if (t < NPTS)      outp[1024 + t] = xn[t];
  else if (t < 64)   outp[1024 + t] = yn[t - 32];
}

// ---------------------------------------------------------------------------
// Phase 2: single block (32x32 threads). Reduce partials -> C, run 100
// Sinkhorn iterations (v-update in-wave, u-update via LDS transpose), emit cost.
// ---------------------------------------------------------------------------
__global__ __launch_bounds__(1024)
void sinkhorn_iterate(const float* __restrict__ ws, float* __restrict__ out) {
  __shared__ float x2s[NPTS];
  __shared__ float y2s[NPTS];
  __shared__ float u_sh[NPTS];
  __shared__ float mT[NPTS * 33];   // padded transpose tile
  __shared__ float wsum[NPTS];

  const int i = threadIdx.x;        // lane index within wave (row index of C)
  const int j = threadIdx.y;        // wave index (column index of C)

  // deterministic reduction of the 256 per-slice partials
  float s = 0.f;
  for (int p = 0; p < GBLK; ++p)
    s += ws[(size_t)p * PART_STRIDE + i * 32 + j];
  if (j == 0) { float a = 0.f; for (int p = 0; p < GBLK; ++p) a += ws[(size_t)p * PART_STRIDE + 1024 + i]; x2s[i] = a; }
  if (j == 1) { float a = 0.f; for (int p = 0; p < GBLK; ++p) a += ws[(size_t)p * PART_STRIDE + 1056 + i]; y2s[i] = a; }
  if (j == 2) u_sh[i] = 0.f;
  __syncthreads();

  const float Cij   = x2s[i] + y2s[j] - 2.f * s;
  const float log_w = __logf(1.f / 32.f + 1e-8f);   // log_mu == log_nu
  float v = 0.f;                                     // per-column potential

  for (int iter = 0; iter < MAX_ITER_S; ++iter) {
    const float ui = u_sh[i];
    // ---- v update: logsumexp over i (within wave) of M(u_old, v_old)
    float mm  = (ui + v - Cij) * INV_EPS;
    float mx  = wave_reduce_max(mm);
    float se  = wave_reduce_sum(__expf(mm - mx));
    float lse = mx + __logf(se);
    v = EPS_F * (log_w - lse) + v;

    // ---- u update: logsumexp over j of M(u_old, v_new) via LDS transpose
    float m2 = (ui + v - Cij) * INV_EPS;
    mT[i * 33 + j] = m2;
    __syncthreads();
    const float uold = u_sh[j];
    float q    = mT[j * 33 + i];          // wave j reduces row i=j over columns
    float qmx  = wave_reduce_max(q);
    float qse  = wave_reduce_sum(__expf(q - qmx));
    float unew = EPS_F * (log_w - (qmx + __logf(qse))) + uold;
    if (i == 0) u_sh[j] = unew;
    __syncthreads();
  }

  // cost = sum(exp(M(u,v)) * C)
  float pi_  = __expf((u_sh[i] + v - Cij) * INV_EPS);
  float term = wave_reduce_sum(pi_ * Cij);
  if (i == 0) wsum[j] = term;
  __syncthreads();
  if (j == 0) {
    float tot = wave_reduce_sum(wsum[i]);
    if (i == 0) out[0] = tot;
  }
}

extern "C" void kernel_launch(void* const* d_in, const int* in_sizes, int n_in,
                              void* d_out, int out_size, void* d_ws, size_t ws_size,
                              hipStream_t stream) {
  const float* x = (const float*)d_in[0];
  const float* y = (const float*)d_in[1];
  float* out = (float*)d_out;
  float* ws  = (float*)d_ws;     // needs GBLK * PART_STRIDE * 4 = ~1.1 MB

  sinkhorn_cost_partial<<<GBLK, 256, 0, stream>>>(x, y, ws);
  sinkhorn_iterate<<<1, dim3(32, 32), 0, stream>>>(ws, out);
}